// NavierStokesPINOLoss_43782896616006
// MI455X (gfx1250) — compile-verified
//
#include <hip/hip_runtime.h>
#include <hip/hip_bf16.h>
#include <stdint.h>

#define W      512
#define TY     8
#define HROWS  (TY + 2)          // tile rows + 2 halo rows
#define NWG_Y  (W / TY)          // 64 y-tiles
#define NBATCH 32                // 4*8 flattened
#define NWG    (NWG_Y * NBATCH)  // 2048 workgroups
#define NPTS_F 8388608.0f        // 32*512*512 (mean denominator)

// gfx1250 async global->LDS builtins (verified present on this toolchain).
#if defined(__HIP_DEVICE_COMPILE__) && __has_builtin(__builtin_amdgcn_global_load_async_to_lds_b128)
#define USE_ASYNC_LDS 1
#else
#define USE_ASYNC_LDS 0
#endif

typedef int   v4i __attribute__((vector_size(16)));
typedef float f4  __attribute__((ext_vector_type(4)));
typedef __attribute__((address_space(1))) v4i gv4i;   // global 16B vector
typedef __attribute__((address_space(3))) v4i lv4i;   // LDS 16B vector

__device__ __forceinline__ void async_copy16(const float* gsrc, float* ldst) {
#if USE_ASYNC_LDS
    // global -> LDS, 16 bytes per lane, tracked by ASYNCcnt
    __builtin_amdgcn_global_load_async_to_lds_b128(
        (gv4i*)(uintptr_t)gsrc,
        (lv4i*)(uint32_t)(uintptr_t)ldst,
        0, 0);
#else
    *(f4*)ldst = *(const f4*)gsrc;
#endif
}

__device__ __forceinline__ void wait_async_lds() {
#if USE_ASYNC_LDS
 #if __has_builtin(__builtin_amdgcn_s_wait_asynccnt)
    __builtin_amdgcn_s_wait_asynccnt(0);
 #else
    asm volatile("s_wait_asynccnt 0" ::: "memory");
 #endif
#endif
}

extern "C" __global__ __launch_bounds__(256)
void ns_pino_main(const float* __restrict__ u_pred,
                  const float* __restrict__ u_prev,
                  float* __restrict__ ws)
{
    __shared__ __align__(16) float su[HROWS * W];   // u_pred ch0, rows y0-1..y0+8
    __shared__ __align__(16) float sv[HROWS * W];   // u_pred ch1
    __shared__ float s_wave[8][2];

    const int t     = threadIdx.x;
    const int tileY = blockIdx.x;           // 0..63
    const int b     = blockIdx.y;           // 0..31
    const int y0    = tileY * TY;
    const size_t baseB = (size_t)b * 2u * W * W;

    // ---- stage u_pred tile (+halo) into LDS: 2 ch * 10 rows * 512 f32 = 2560 x 16B ----
    const int SEGS = W / 4;                  // 128 16B segments per row
    #pragma unroll
    for (int k = 0; k < (2 * HROWS * SEGS) / 256; ++k) {   // 10 iterations
        int chunk = t + k * 256;
        int ch    = chunk / (HROWS * SEGS);
        int rem   = chunk - ch * (HROWS * SEGS);
        int row   = rem / SEGS;
        int seg   = rem - row * SEGS;
        int gy    = (y0 - 1 + row) & (W - 1);             // periodic wrap
        const float* gsrc = u_pred + baseB + (size_t)ch * (W * W)
                                   + (size_t)gy * W + seg * 4;
        float* ldst = (ch ? sv : su) + row * W + seg * 4;
        async_copy16(gsrc, ldst);
    }
    wait_async_lds();
    __syncthreads();

    // ---- stencil + residuals: 4 groups/thread, 4 consecutive x per group ----
    const float* prev0 = u_prev + baseB;
    const float* prev1 = u_prev + baseB + (size_t)(W * W);

    float accP = 0.0f, accD = 0.0f;
    #pragma unroll
    for (int k = 0; k < 4; ++k) {
        int gidx = t + k * 256;               // 0..1023 float4-groups in tile
        int r    = gidx >> 7;                 // 128 groups per row
        int xq   = (gidx & 127) << 2;         // first x of the group
        int lr   = r + 1;
        int y    = y0 + r;
        int rowc = lr * W;

        // center rows (b128), up/down rows (b128)
        f4 uc4 = *(const f4*)&su[rowc + xq];
        f4 vc4 = *(const f4*)&sv[rowc + xq];
        f4 uup = *(const f4*)&su[rowc - W + xq];
        f4 udn = *(const f4*)&su[rowc + W + xq];
        f4 vup = *(const f4*)&sv[rowc - W + xq];
        f4 vdn = *(const f4*)&sv[rowc + W + xq];
        // periodic row-edge scalars
        int xl = (xq + W - 1) & (W - 1);
        int xr = (xq + 4) & (W - 1);
        float ul = su[rowc + xl], ur = su[rowc + xr];
        float vl = sv[rowc + xl], vr = sv[rowc + xr];

        // u_prev: one non-temporal b128 per channel (streamed once, skip cache)
        const f4 p0 = __builtin_nontemporal_load((const f4*)(prev0 + (size_t)y * W + xq));
        const f4 p1 = __builtin_nontemporal_load((const f4*)(prev1 + (size_t)y * W + xq));

        const float ucA[6] = {ul, uc4.x, uc4.y, uc4.z, uc4.w, ur};
        const float vcA[6] = {vl, vc4.x, vc4.y, vc4.z, vc4.w, vr};
        const float uupA[4] = {uup.x, uup.y, uup.z, uup.w};
        const float udnA[4] = {udn.x, udn.y, udn.z, udn.w};
        const float vupA[4] = {vup.x, vup.y, vup.z, vup.w};
        const float vdnA[4] = {vdn.x, vdn.y, vdn.z, vdn.w};
        const float pA0[4]  = {p0.x, p0.y, p0.z, p0.w};
        const float pA1[4]  = {p1.x, p1.y, p1.z, p1.w};

        #pragma unroll
        for (int j = 0; j < 4; ++j) {
            float uc = ucA[j + 1], vc = vcA[j + 1];
            float u_x = 0.5f * (ucA[j + 2] - ucA[j]);
            float v_x = 0.5f * (vcA[j + 2] - vcA[j]);
            float u_y = 0.5f * (udnA[j] - uupA[j]);
            float v_y = 0.5f * (vdnA[j] - vupA[j]);
            float lap_u = ucA[j + 2] + ucA[j] + udnA[j] + uupA[j] - 4.0f * uc;
            float lap_v = vcA[j + 2] + vcA[j] + vdnA[j] + vupA[j] - 4.0f * vc;

            float du = (uc - pA0[j]) * 100.0f;            // 1/DT
            float dv = (vc - pA1[j]) * 100.0f;

            float rx  = du + uc * u_x + vc * u_y - 0.001f * lap_u;   // NU = 1e-3
            float ry  = dv + uc * v_x + vc * v_y - 0.001f * lap_v;
            float dvg = u_x + v_y;

            accP += rx * rx + ry * ry;
            accD += dvg * dvg;
        }
    }

    // ---- deterministic reduction: wave32 shuffle tree -> per-wave LDS -> thread 0 ----
    #pragma unroll
    for (int off = 16; off > 0; off >>= 1) {
        accP += __shfl_down(accP, off, 32);
        accD += __shfl_down(accD, off, 32);
    }
    int wave = t >> 5, lane = t & 31;
    if (lane == 0) { s_wave[wave][0] = accP; s_wave[wave][1] = accD; }
    __syncthreads();
    if (t == 0) {
        float sp = 0.0f, sd = 0.0f;
        #pragma unroll
        for (int wv = 0; wv < 8; ++wv) { sp += s_wave[wv][0]; sd += s_wave[wv][1]; }
        int wg = b * NWG_Y + tileY;           // 0..2047
        ws[wg]       = sp;
        ws[NWG + wg] = sd;
    }
}

extern "C" __global__ __launch_bounds__(256)
void ns_pino_finalize(const float* __restrict__ ws, float* __restrict__ out)
{
    __shared__ float r0[256], r1[256];
    int t = threadIdx.x;
    float s0 = 0.0f, s1 = 0.0f;
    for (int i = t; i < NWG; i += 256) { s0 += ws[i]; s1 += ws[NWG + i]; }
    r0[t] = s0; r1[t] = s1;
    __syncthreads();
    #pragma unroll
    for (int off = 128; off > 0; off >>= 1) {
        if (t < off) { r0[t] += r0[t + off]; r1[t] += r1[t + off]; }
        __syncthreads();
    }
    if (t == 0) {
        const float invN = 1.0f / NPTS_F;
        float pde = r0[0] * invN;
        float div = r1[0] * invN;
        out[0] = pde + 0.1f * div;   // physics_loss
        out[1] = pde;                // pde_loss
        out[2] = div;                // div_loss
    }
}

extern "C" void kernel_launch(void* const* d_in, const int* in_sizes, int n_in,
                              void* d_out, int out_size, void* d_ws, size_t ws_size,
                              hipStream_t stream) {
    const float* u_pred = (const float*)d_in[0];
    const float* u_prev = (const float*)d_in[1];
    float* ws  = (float*)d_ws;    // 2*2048 f32 partials = 16 KB
    float* out = (float*)d_out;   // 3 f32 scalars

    dim3 grid(NWG_Y, NBATCH);     // 64 x 32 = 2048 workgroups
    dim3 block(256);
    ns_pino_main<<<grid, block, 0, stream>>>(u_pred, u_prev, ws);
    ns_pino_finalize<<<1, 256, 0, stream>>>(ws, out);
}